// Agreement_Routing_Down_1580547966733
// MI455X (gfx1250) — compile-verified
//
#include <hip/hip_runtime.h>

typedef __attribute__((ext_vector_type(2))) float v2f;
typedef __attribute__((ext_vector_type(4))) float v4f;
typedef __attribute__((ext_vector_type(8))) float v8f;

namespace {
constexpr int kB = 8, kO = 32, kH = 14, kW = 14, kI = 32, kD = 16;
constexpr int kOHW   = kO * kH * kW;          // 6272
constexpr int kSites = kB * kOHW;             // 50176
constexpr int kWavesPerBlock = 8;             // 256 threads = 8 wave32
constexpr int kRowStride = 20;                // 16 + 4 pad floats: conflict-free transpose
constexpr float kEps = 1e-7f;
constexpr int kIters = 3;
}

// single-instruction lane-XOR exchange: ds_swizzle_b32, group-of-32 mode,
// and_mask=0x1f, or_mask=0, xor_mask=MASK  ->  offset = 0x1f | (MASK<<10)
template <int IMM>
__device__ __forceinline__ float swz(float x) {
    return __builtin_bit_cast(float,
        __builtin_amdgcn_ds_swizzle(__builtin_bit_cast(int, x), IMM));
}
#define SWZ_X16 0x401f
#define SWZ_X8  0x201f
#define SWZ_X4  0x101f
#define SWZ_X2  0x081f
#define SWZ_X1  0x041f

__global__ __launch_bounds__(256)
void caps_route_kernel(const float* __restrict__ preds,
                       const float* __restrict__ blog,
                       float* __restrict__ out)
{
    __shared__ float lds[kWavesPerBlock * kI * kRowStride];   // 20 KB

    const int wave = threadIdx.x >> 5;
    const int tid  = threadIdx.x & 31;
    const int hi   = tid >> 4;          // 0: lanes 0-15, 1: lanes 16-31
    const int m    = tid & 15;
    const int site = blockIdx.x * kWavesPerBlock + wave;      // grid divides exactly

    // each wave uses ONLY its own LDS slice; DS ops of a wave execute in order,
    // so no block barriers are needed anywhere in this kernel.
    float* wlds = lds + wave * (kI * kRowStride);

    // ---- read row `tid`, quads ROTATED by lane half: col = (4j + 8*hi) & 15 ----
    // so Prow[0..7] pairs this lane's v-half and Prow[8..15] the other: no cndmasks.
    const float* tile = preds + (size_t)site * (kI * kD);
    alignas(16) float Prow[16];
#pragma unroll
    for (int j = 0; j < 4; ++j) {
        const int col = (4 * j + 8 * hi) & 15;
        *(float4*)&Prow[4 * j] = *(const float4*)(tile + tid * kD + col);
    }

    // routing logits: b is [1,O,H,W,I], broadcast over batch
    float bb = blog[(site % kOHW) * kI + tid];

    // ---- LDS transpose (store quads at their TRUE column, undoing the rotation) ----
#pragma unroll
    for (int j = 0; j < 4; ++j) {
        const int col = (4 * j + 8 * hi) & 15;
        *(float4*)&wlds[tid * kRowStride + col] = *(const float4*)&Prow[4 * j];
    }

    // agreement operands: Qa = Prow[0..7] (pairs own v-half); QbX = PARTNER lane's
    // other-half row data (one-time swizzle) so each lane computes the partial its
    // partner needs -> one swizzle per round instead of 8.
    float Qa[8], QbX[8];
#pragma unroll
    for (int r = 0; r < 8; ++r) {
        Qa[r]  = Prow[r];
        QbX[r] = swz<SWZ_X16>(Prow[8 + r]);
        asm volatile("" : "+v"(Qa[r]), "+v"(QbX[r]));
    }

    // WMMA K-chunk c covers rows (2c, 2c+1, 16+2c, 17+2c): chunk order is free, and
    // this choice makes the B-operand e-pairs CONTIGUOUS under the identity e-layout
    // (lanes<16 use e[2c],e[2c+1]; lanes>=16 use e[16+2c],e[17+2c]) -> plain wlds[tid]
    // store, 4x broadcast ds_load_b128 per round, zero per-round address math.
    // A 16x4 f32 layout: VGPR0 = K-row0 (lanes<16) / K-row2 (lanes>=16), VGPR1 = row1/row3.
    v2f PA[8];
#pragma unroll
    for (int c = 0; c < 8; ++c) {
        const int r = 2 * c + 16 * hi;
        v2f t;
        t.x = wlds[r * kRowStride + m];
        t.y = wlds[(r + 1) * kRowStride + m];
        PA[c] = t;
        asm volatile("" : "+v"(PA[c]));
    }

    const float* ebase = wlds + 16 * hi;

    alignas(16) float vr[8]; // lanes<16 hold v[0..7], lanes>=16 hold v[8..15]
    float u = 0.f;           // partial agreement computed FOR the partner lane

#pragma unroll
    for (int it = 0; it <= kIters; ++it) {
        if (it) {
            // agreement a_i = P[i,:].v = own-half dot + partner-computed other-half dot
            float a = 0.f;
#pragma unroll
            for (int r = 0; r < 8; ++r) a = fmaf(vr[r], Qa[r], a);
            a += swz<SWZ_X16>(u);
            bb += a;
        }

        // softmax over I = 32 (one logit per lane). No max-shift needed: logits start
        // at 0 and |agreement| <= ||P_row||*||v|| with ||v||<1 -> |bb| ~ 25, no overflow.
        // Store unnormalized e; 1/S folds into the squash factor.
        const float e = __expf(bb);
        wlds[tid] = e;                                  // ds_store_b32, wave-private
        float S = e;
        S += swz<SWZ_X16>(S);
        S += swz<SWZ_X8>(S);
        S += swz<SWZ_X4>(S);
        S += swz<SWZ_X2>(S);
        S += swz<SWZ_X1>(S);

        // t = P^T e via 8 chained f32 WMMAs (K = 32 total = I); B pairs come two
        // chunks at a time from one broadcast ds_load_b128.
        v8f acc = {0.f, 0.f, 0.f, 0.f, 0.f, 0.f, 0.f, 0.f};
#pragma unroll
        for (int c2 = 0; c2 < 4; ++c2) {
            const v4f q = *(const v4f*)&ebase[4 * c2];
            const v2f bv0 = __builtin_shufflevector(q, q, 0, 1);
            const v2f bv1 = __builtin_shufflevector(q, q, 2, 3);
            acc = __builtin_amdgcn_wmma_f32_16x16x4_f32(
                false, PA[2 * c2],     false, bv0, (short)0, acc, false, false);
            acc = __builtin_amdgcn_wmma_f32_16x16x4_f32(
                false, PA[2 * c2 + 1], false, bv1, (short)0, acc, false, false);
        }
        // D layout: lanes<16 VGPR r = t[r]; lanes>=16 VGPR r = t[8+r];  s = t / S

        // squash with 1/S folded in: sq = |t|^2/S^2, v = [sq/((1+sq)sqrt(sq+eps))]/S * t
        float p = 0.f;
#pragma unroll
        for (int r = 0; r < 8; ++r) p = fmaf(acc[r], acc[r], p);
        const float invS = __builtin_amdgcn_rcpf(S);
        const float sq   = (p + swz<SWZ_X16>(p)) * invS * invS;
        const float f    = sq * __builtin_amdgcn_rcpf(1.f + sq)
                              * __builtin_amdgcn_rsqf(sq + kEps) * invS;
#pragma unroll
        for (int r = 0; r < 8; ++r) vr[r] = f * acc[r];

        if (it < kIters) {
            // partial the PARTNER needs next round: its missing v-half dotted with
            // its row's other-half columns (held here as QbX)
            float uu = 0.f;
#pragma unroll
            for (int r = 0; r < 8; ++r) uu = fmaf(vr[r], QbX[r], uu);
            u = uu;
        }
    }

    // ---- write v[16]: lane0 has v[0..7], lane16 has v[8..15] -> 2x b128 each ----
    if (m == 0) {
        float* op = out + (size_t)site * kD + hi * 8;
        *(float4*)(op)     = *(const float4*)&vr[0];
        *(float4*)(op + 4) = *(const float4*)&vr[4];
    }
}

extern "C" void kernel_launch(void* const* d_in, const int* in_sizes, int n_in,
                              void* d_out, int out_size, void* d_ws, size_t ws_size,
                              hipStream_t stream) {
    const float* preds = (const float*)d_in[0];   // [B,O,H,W,I,D] f32
    const float* blog  = (const float*)d_in[1];   // [1,O,H,W,I]   f32
    float* out = (float*)d_out;                   // [B,O,H,W,D]   f32

    dim3 grid(kSites / kWavesPerBlock);           // 6272 blocks, exact
    caps_route_kernel<<<grid, 256, 0, stream>>>(preds, blog, out);
}